// GCN1110_20469814133405
// MI455X (gfx1250) — compile-verified
//
#include <hip/hip_runtime.h>
#include <math.h>

typedef float v2f __attribute__((ext_vector_type(2)));
typedef float v8f __attribute__((ext_vector_type(8)));

static inline int cdiv_i(long long a, long long b) { return (int)((a + b - 1) / b); }

// ---------------------------------------------------------------------------
// Utility kernels
// ---------------------------------------------------------------------------
__global__ void zero_f32(float* __restrict__ p, long long n) {
  long long i = (long long)blockIdx.x * blockDim.x + threadIdx.x;
  long long stride = (long long)gridDim.x * blockDim.x;
  for (; i < n; i += stride) p[i] = 0.0f;
}

__global__ void relu_f32(float* __restrict__ p, long long n) {
  long long i = (long long)blockIdx.x * blockDim.x + threadIdx.x;
  long long stride = (long long)gridDim.x * blockDim.x;
  for (; i < n; i += stride) p[i] = fmaxf(p[i], 0.0f);
}

__global__ void deg_count(const int* __restrict__ dst, int E, float* __restrict__ deg) {
  int i = blockIdx.x * blockDim.x + threadIdx.x;
  if (i < E) atomicAdd(&deg[dst[i]], 1.0f);
}

__global__ void deg_to_dinv(float* __restrict__ deg, long long n) {
  long long i = (long long)blockIdx.x * blockDim.x + threadIdx.x;
  long long stride = (long long)gridDim.x * blockDim.x;
  for (; i < n; i += stride) {
    float d = deg[i];
    deg[i] = (d > 0.0f) ? rsqrtf(d) : 0.0f;
  }
}

// ---------------------------------------------------------------------------
// fp32 WMMA GEMM:  D[M x Ncols] = A[M x K] @ B[K x Ncols] (+ optional bias)
// One block per 16-row strip of A/D; Ncols/16 waves per block, one 16x16 D
// tile per wave. A strip staged in LDS with stride K+1 (bank-conflict-free:
// K % 64 == 0 for all call sites). K must be a multiple of 4.
// ---------------------------------------------------------------------------
__global__ void gemm_wmma_f32(const float* __restrict__ A, const float* __restrict__ B,
                              const float* __restrict__ bias, float* __restrict__ D,
                              int M, int K, int Ncols) {
  extern __shared__ float As[];                 // 16 * (K+1) floats
  const int row0 = blockIdx.x * 16;
  const int tid  = threadIdx.x;
  const int ldA  = K + 1;

  // Cooperative, coalesced load of the 16 x K strip of A into LDS.
  for (int idx = tid; idx < 16 * K; idx += blockDim.x) {
    int r = idx / K;
    int k = idx - r * K;
    int row = row0 + r;
    if (row >= M) row = M - 1;                  // clamp (M is a multiple of 16 here anyway)
    As[r * ldA + k] = A[(long long)row * K + k];
  }
  __syncthreads();

  const int wave = tid >> 5;
  const int lane = tid & 31;
  const int half = lane >> 4;                   // 0: lanes 0-15, 1: lanes 16-31
  const int mn   = lane & 15;                   // A row / B,D column within tile
  const int kb   = half * 2;                    // k sub-offset of this half-wave
  const int col0 = wave * 16;

  v8f acc = {};
  #pragma unroll 4
  for (int kk = 0; kk < K; kk += 4) {
    // A fragment: 16x4 f32. lane(l): m = l&15, holds K = (l>>4)*2 + {0,1}
    v2f a, b;
    a.x = As[mn * ldA + kk + kb];
    a.y = As[mn * ldA + kk + kb + 1];
    // B fragment: 4x16 f32. lane(l): n = l&15, holds K = (l>>4)*2 + {0,1}
    const float* Bp = B + (long long)(kk + kb) * Ncols + col0 + mn;
    b.x = Bp[0];
    b.y = Bp[Ncols];
    acc = __builtin_amdgcn_wmma_f32_16x16x4_f32(
        /*neg_a=*/false, a, /*neg_b=*/false, b,
        /*c_mod=*/(short)0, acc, /*reuse_a=*/false, /*reuse_b=*/false);
  }

  const float bv = bias ? bias[col0 + mn] : 0.0f;
  // D tile: VGPR r -> M = r + 8*half, N = mn
  #pragma unroll
  for (int r = 0; r < 8; ++r) {
    int row = row0 + r + half * 8;
    if (row < M) D[(long long)row * Ncols + col0 + mn] = acc[r] + bv;
  }
}

// ---------------------------------------------------------------------------
// Edge-normalized scatter-add: out[dst] += dinv[src]*dinv[dst] * HH[src]
// One thread per (edge, feature) element, grid-stride.
// ---------------------------------------------------------------------------
__global__ void gcn_aggregate(const int* __restrict__ src, const int* __restrict__ dst, int E,
                              const float* __restrict__ dinv,
                              const float* __restrict__ HH, int ldH,
                              float* __restrict__ out, int ldOut, int width) {
  long long total  = (long long)E * width;
  long long stride = (long long)gridDim.x * blockDim.x;
  for (long long i = (long long)blockIdx.x * blockDim.x + threadIdx.x; i < total; i += stride) {
    int e = (int)(i / width);
    int h = (int)(i - (long long)e * width);
    int s = src[e];
    int d = dst[e];
    float nrm = dinv[s] * dinv[d];
    atomicAdd(&out[(long long)d * ldOut + h], HH[(long long)s * ldH + h] * nrm);
  }
}

// R1 init: columns [0,384) = b1 (atomics accumulate on top),
//          columns [384,512) = HH1 + b1 (edge_id branch, norm == 1).
__global__ void build_R1(const float* __restrict__ HH1, const float* __restrict__ b1,
                         float* __restrict__ R1, int N) {
  long long total  = (long long)N * 512;
  long long stride = (long long)gridDim.x * blockDim.x;
  for (long long i = (long long)blockIdx.x * blockDim.x + threadIdx.x; i < total; i += stride) {
    int col = (int)(i & 511);
    long long n = i >> 9;
    int br = col >> 7;
    int h  = col & 127;
    float v = b1[h];
    if (br == 3) v += HH1[n * 128 + h];
    R1[i] = v;
  }
}

// R2 init: columns [0,192) = b2, columns [192,256) = HH2 + b2 (edge_id branch).
__global__ void build_R2(const float* __restrict__ HH2, const float* __restrict__ b2,
                         float* __restrict__ R2, int N) {
  long long total  = (long long)N * 256;
  long long stride = (long long)gridDim.x * blockDim.x;
  for (long long i = (long long)blockIdx.x * blockDim.x + threadIdx.x; i < total; i += stride) {
    int col = (int)(i & 255);
    long long n = i >> 8;
    int br = col >> 6;
    int h  = col & 63;
    float v = b2[h];
    if (br == 3) v += HH2[n * 64 + h];
    R2[i] = v;
  }
}

// In-place row-wise log-softmax over 64 columns; one wave32 per row (2 cols/lane).
__global__ void log_softmax64(float* __restrict__ out, int N) {
  int row  = blockIdx.x * (blockDim.x >> 5) + (threadIdx.x >> 5);
  int lane = threadIdx.x & 31;
  if (row >= N) return;
  float* p = out + (long long)row * 64;
  float a = p[lane];
  float b = p[lane + 32];
  float mx = fmaxf(a, b);
  #pragma unroll
  for (int off = 16; off > 0; off >>= 1) mx = fmaxf(mx, __shfl_xor(mx, off));
  float s = __expf(a - mx) + __expf(b - mx);
  #pragma unroll
  for (int off = 16; off > 0; off >>= 1) s += __shfl_xor(s, off);
  float lse = mx + __logf(s);
  p[lane]      = a - lse;
  p[lane + 32] = b - lse;
}

// ---------------------------------------------------------------------------
// Orchestration
// ---------------------------------------------------------------------------
extern "C" void kernel_launch(void* const* d_in, const int* in_sizes, int n_in,
                              void* d_out, int out_size, void* d_ws, size_t ws_size,
                              hipStream_t stream) {
  const float* x  = (const float*)d_in[0];
  const int* e1   = (const int*)d_in[1];   // edge_index (2 x E1)
  const int* e2   = (const int*)d_in[2];   // edge_2hop  (2 x E2)
  const int* e3   = (const int*)d_in[3];   // edge_knn   (2 x E3)
  // d_in[4] = edge_id: pure self-loops -> handled analytically (out = hh + b)
  const float* W1 = (const float*)d_in[5];
  const float* b1 = (const float*)d_in[6];
  const float* W2 = (const float*)d_in[7];
  const float* b2 = (const float*)d_in[8];
  const float* Wl = (const float*)d_in[9];
  const float* bl = (const float*)d_in[10];

  const int F = 512, H = 128, C = 64;
  const int N  = in_sizes[0] / F;
  const int E1 = in_sizes[1] / 2;
  const int E2 = in_sizes[2] / 2;
  const int E3 = in_sizes[3] / 2;

  // Workspace layout (floats): deg/dinv x3 | HH1 | R1 | HH2 | R2
  float* ws   = (float*)d_ws;
  float* dinv = ws;                                   // 3*N
  float* HH1  = dinv + 3LL * N;                       // N*H
  float* R1   = HH1 + (long long)N * H;               // N*4H
  float* HH2  = R1 + (long long)N * 4 * H;            // N*C
  float* R2   = HH2 + (long long)N * C;               // N*4C

  const int T = 256;
  const int AGG_MAX_BLOCKS = 131072;

  // --- degrees -> D^-1/2 (shared by both layers) ---
  zero_f32<<<cdiv_i(3LL * N, T), T, 0, stream>>>(dinv, 3LL * N);
  deg_count<<<cdiv_i(E1, T), T, 0, stream>>>(e1 + E1, E1, dinv);
  deg_count<<<cdiv_i(E2, T), T, 0, stream>>>(e2 + E2, E2, dinv + N);
  deg_count<<<cdiv_i(E3, T), T, 0, stream>>>(e3 + E3, E3, dinv + 2LL * N);
  deg_to_dinv<<<cdiv_i(3LL * N, T), T, 0, stream>>>(dinv, 3LL * N);

  const int mtiles = cdiv_i(N, 16);

  // --- Layer 1: HH1 = x @ W1 (shared by all 4 branches) ---
  gemm_wmma_f32<<<mtiles, 32 * (H / 16), (size_t)16 * (F + 1) * sizeof(float), stream>>>(
      x, W1, nullptr, HH1, N, F, H);

  build_R1<<<cdiv_i((long long)N * 4 * H, T), T, 0, stream>>>(HH1, b1, R1, N);

  {
    long long t1 = (long long)E1 * H, t2 = (long long)E2 * H, t3 = (long long)E3 * H;
    int g1 = min(cdiv_i(t1, T), AGG_MAX_BLOCKS);
    int g2 = min(cdiv_i(t2, T), AGG_MAX_BLOCKS);
    int g3 = min(cdiv_i(t3, T), AGG_MAX_BLOCKS);
    gcn_aggregate<<<g1, T, 0, stream>>>(e1, e1 + E1, E1, dinv,           HH1, H, R1,           4 * H, H);
    gcn_aggregate<<<g2, T, 0, stream>>>(e2, e2 + E2, E2, dinv + N,       HH1, H, R1 + H,       4 * H, H);
    gcn_aggregate<<<g3, T, 0, stream>>>(e3, e3 + E3, E3, dinv + 2LL * N, HH1, H, R1 + 2 * H,   4 * H, H);
  }

  relu_f32<<<cdiv_i((long long)N * 4 * H, T), T, 0, stream>>>(R1, (long long)N * 4 * H);

  // --- Layer 2: HH2 = R1 @ W2 (shared by all 4 branches) ---
  gemm_wmma_f32<<<mtiles, 32 * (C / 16), (size_t)16 * (4 * H + 1) * sizeof(float), stream>>>(
      R1, W2, nullptr, HH2, N, 4 * H, C);

  build_R2<<<cdiv_i((long long)N * 4 * C, T), T, 0, stream>>>(HH2, b2, R2, N);

  {
    long long t1 = (long long)E1 * C, t2 = (long long)E2 * C, t3 = (long long)E3 * C;
    int g1 = min(cdiv_i(t1, T), AGG_MAX_BLOCKS);
    int g2 = min(cdiv_i(t2, T), AGG_MAX_BLOCKS);
    int g3 = min(cdiv_i(t3, T), AGG_MAX_BLOCKS);
    gcn_aggregate<<<g1, T, 0, stream>>>(e1, e1 + E1, E1, dinv,           HH2, C, R2,           4 * C, C);
    gcn_aggregate<<<g2, T, 0, stream>>>(e2, e2 + E2, E2, dinv + N,       HH2, C, R2 + C,       4 * C, C);
    gcn_aggregate<<<g3, T, 0, stream>>>(e3, e3 + E3, E3, dinv + 2LL * N, HH2, C, R2 + 2 * C,   4 * C, C);
  }

  // --- Output head: logits = R2 @ Wlin + blin, then in-place log-softmax ---
  float* logits = (float*)d_out;
  gemm_wmma_f32<<<mtiles, 32 * (C / 16), (size_t)16 * (4 * C + 1) * sizeof(float), stream>>>(
      R2, Wl, bl, logits, N, 4 * C, C);

  log_softmax64<<<cdiv_i(N, T / 32), T, 0, stream>>>(logits, N);
}